// QGNNGraphClassifier_44272522887298
// MI455X (gfx1250) — compile-verified
//
#include <hip/hip_runtime.h>
#include <hip/hip_bf16.h>
#include <math.h>

// ---------------------------------------------------------------------------
// QGNN forward for MI455X (gfx1250, wave32).
//  K1: node MLP  8->128 (WMMA f32 16x16x4) ->lrelu-> 128->2 ->tanh*pi  -> x, x_acc
//  K1: edge MLP  4->128 (WMMA f32 16x16x4) ->lrelu-> 128->2 ->tanh*pi  -> e
//  K2: per-node 9-qubit statevector sim, one wave32 per node, state in VGPRs
//      (16 complex amps / lane). wires 0..4 = lane bits (ds_swizzle shfl_xor),
//      wires 5..8 = register bits (local FMA only). Weight-derived gate
//      matrices hoisted + native __sinf/__cosf (angles bounded by pi, no
//      argument reduction needed). Then update MLP + atomic scatter.
//  K3: segment mean pool (LDS atomics) + 2->2->2 head.
// ---------------------------------------------------------------------------

typedef __attribute__((ext_vector_type(2))) float v2f;
typedef __attribute__((ext_vector_type(8))) float v8f;

#define LRELU(v) ((v) > 0.0f ? (v) : 0.01f * (v))
static __device__ __forceinline__ float sx(float v, int m) { return __shfl_xor(v, m, 32); }

// ======================= feature MLP (WMMA f32) ============================
template <int KIN>
__global__ void feat_mlp_kernel(const float* __restrict__ feat,
                                const float* __restrict__ W1, const float* __restrict__ b1,
                                const float* __restrict__ W2, const float* __restrict__ b2,
                                float* __restrict__ out0, float* __restrict__ out1,
                                int nrows) {
  __shared__ float hbuf[4][16][128];  // 32 KB
  const int lane = threadIdx.x & 31;
  const int wave = threadIdx.x >> 5;
  const int hl   = lane >> 4;
  const int mloc = lane & 15;
  const int rowBase = blockIdx.x * 64 + wave * 16;

  v2f afrag[KIN / 4];
#pragma unroll
  for (int ks = 0; ks < KIN / 4; ++ks) {
    const int m = rowBase + mloc;
    const int k0 = ks * 4 + 2 * hl;
    afrag[ks].x = feat[m * KIN + k0];
    afrag[ks].y = feat[m * KIN + k0 + 1];
  }

#pragma unroll
  for (int nt = 0; nt < 8; ++nt) {
    v8f acc = {};
#pragma unroll
    for (int ks = 0; ks < KIN / 4; ++ks) {
      const int k0 = ks * 4 + 2 * hl;
      v2f bfrag;
      bfrag.x = W1[(k0)     * 128 + nt * 16 + mloc];
      bfrag.y = W1[(k0 + 1) * 128 + nt * 16 + mloc];
      acc = __builtin_amdgcn_wmma_f32_16x16x4_f32(false, afrag[ks], false, bfrag,
                                                  (short)0, acc, false, false);
    }
#pragma unroll
    for (int r = 0; r < 8; ++r) {
      const int mrow = r + 8 * hl;
      const int n = nt * 16 + mloc;
      float v = acc[r] + b1[n];
      hbuf[wave][mrow][n] = LRELU(v);
    }
  }
  __syncthreads();

  float s0 = 0.f, s1 = 0.f;
  const int n0 = 64 * hl;
#pragma unroll 8
  for (int n = 0; n < 64; ++n) {
    const float v = hbuf[wave][mloc][n0 + n];
    s0 = fmaf(v, W2[(n0 + n) * 2 + 0], s0);
    s1 = fmaf(v, W2[(n0 + n) * 2 + 1], s1);
  }
  s0 += sx(s0, 16);
  s1 += sx(s1, 16);
  const int row = rowBase + mloc;
  if (hl == 0 && row < nrows) {
    const float PIF = 3.14159265358979323846f;
    const float o0 = tanhf(s0 + b2[0]) * PIF;
    const float o1 = tanhf(s1 + b2[1]) * PIF;
    out0[row * 2] = o0;
    out0[row * 2 + 1] = o1;
    if (out1) { out1[row * 2] = o0; out1[row * 2 + 1] = o1; }
  }
}

// ======================= 9-qubit statevector sim ===========================
// amplitude idx[8:0]; wire w <-> bit (8-w). lane = idx>>4, r = idx&15.
// wires 0..4 -> lane bits 4..0 ; wires 5..8 -> register bits 3..0.
struct Cx { float re, im; };
struct U2 { Cx u00, u01, u10, u11; };

static __device__ __forceinline__ Cx cmul(Cx a, Cx b) {
  return { fmaf(a.re, b.re, -a.im * b.im), fmaf(a.re, b.im, a.im * b.re) };
}
static __device__ __forceinline__ Cx cfma2(Cx A, Cx x, Cx B, Cx y) {  // A*x + B*y
  float re = fmaf(A.re, x.re, fmaf(-A.im, x.im, fmaf(B.re, y.re, -B.im * y.im)));
  float im = fmaf(A.re, x.im, fmaf(A.im, x.re, fmaf(B.re, y.im, B.im * y.re)));
  return { re, im };
}

template <int W>
static __device__ __forceinline__ void apply1(Cx* s, const U2& U, int lane) {
  if constexpr (W >= 5) {
    constexpr int rb = 1 << (8 - W);
#pragma unroll
    for (int r0 = 0; r0 < 16; ++r0)
      if ((r0 & rb) == 0) {
        const int r1 = r0 | rb;
        Cx a0 = s[r0], a1 = s[r1];
        s[r0] = cfma2(U.u00, a0, U.u01, a1);
        s[r1] = cfma2(U.u10, a0, U.u11, a1);
      }
  } else {
    constexpr int lm = 1 << (4 - W);
    const bool hi = (lane & lm) != 0;
    Cx A = hi ? U.u10 : U.u00;
    Cx B = hi ? U.u11 : U.u01;
#pragma unroll
    for (int r = 0; r < 16; ++r) {
      Cx p = { sx(s[r].re, lm), sx(s[r].im, lm) };
      Cx lo = hi ? p : s[r];
      Cx hv = hi ? s[r] : p;
      s[r] = cfma2(A, lo, B, hv);
    }
  }
}

template <int CW, int TW>
static __device__ __forceinline__ void applyC(Cx* s, const U2& U, int lane) {
  if constexpr (CW >= 5 && TW >= 5) {
    constexpr int cb = 1 << (8 - CW), tb = 1 << (8 - TW);
#pragma unroll
    for (int r0 = 0; r0 < 16; ++r0)
      if ((r0 & tb) == 0 && (r0 & cb) != 0) {
        const int r1 = r0 | tb;
        Cx a0 = s[r0], a1 = s[r1];
        s[r0] = cfma2(U.u00, a0, U.u01, a1);
        s[r1] = cfma2(U.u10, a0, U.u11, a1);
      }
  } else if constexpr (CW >= 5 && TW < 5) {
    constexpr int cb = 1 << (8 - CW), lm = 1 << (4 - TW);
    const bool hi = (lane & lm) != 0;
    Cx A = hi ? U.u10 : U.u00;
    Cx B = hi ? U.u11 : U.u01;
#pragma unroll
    for (int r = 0; r < 16; ++r)
      if (r & cb) {  // compile-time per unrolled r: all lanes execute the shfl
        Cx p = { sx(s[r].re, lm), sx(s[r].im, lm) };
        Cx lo = hi ? p : s[r];
        Cx hv = hi ? s[r] : p;
        s[r] = cfma2(A, lo, B, hv);
      }
  } else if constexpr (CW < 5 && TW >= 5) {
    constexpr int cm = 1 << (4 - CW), tb = 1 << (8 - TW);
    const bool c = (lane & cm) != 0;
#pragma unroll
    for (int r0 = 0; r0 < 16; ++r0)
      if ((r0 & tb) == 0) {
        const int r1 = r0 | tb;
        Cx a0 = s[r0], a1 = s[r1];
        Cx n0 = cfma2(U.u00, a0, U.u01, a1);
        Cx n1 = cfma2(U.u10, a0, U.u11, a1);
        s[r0].re = c ? n0.re : a0.re; s[r0].im = c ? n0.im : a0.im;
        s[r1].re = c ? n1.re : a1.re; s[r1].im = c ? n1.im : a1.im;
      }
  } else {
    constexpr int cm = 1 << (4 - CW), lm = 1 << (4 - TW);
    const bool c = (lane & cm) != 0;
    const bool hi = (lane & lm) != 0;
    Cx A = hi ? U.u10 : U.u00;
    Cx B = hi ? U.u11 : U.u01;
#pragma unroll
    for (int r = 0; r < 16; ++r) {
      Cx p = { sx(s[r].re, lm), sx(s[r].im, lm) };
      Cx lo = hi ? p : s[r];
      Cx hv = hi ? s[r] : p;
      Cx n = cfma2(A, lo, B, hv);
      s[r].re = c ? n.re : s[r].re;
      s[r].im = c ? n.im : s[r].im;
    }
  }
}

template <int CW, int TW>
static __device__ __forceinline__ void cnotg(Cx* s, int lane) {
  if constexpr (CW >= 5 && TW >= 5) {
    constexpr int cb = 1 << (8 - CW), tb = 1 << (8 - TW);
#pragma unroll
    for (int r0 = 0; r0 < 16; ++r0)
      if ((r0 & tb) == 0 && (r0 & cb) != 0) {
        const int r1 = r0 | tb;
        Cx t = s[r0]; s[r0] = s[r1]; s[r1] = t;
      }
  } else if constexpr (CW >= 5 && TW < 5) {
    constexpr int cb = 1 << (8 - CW), lm = 1 << (4 - TW);
#pragma unroll
    for (int r = 0; r < 16; ++r)
      if (r & cb) { s[r].re = sx(s[r].re, lm); s[r].im = sx(s[r].im, lm); }
  } else if constexpr (CW < 5 && TW >= 5) {
    constexpr int cm = 1 << (4 - CW), tb = 1 << (8 - TW);
    const bool c = (lane & cm) != 0;
#pragma unroll
    for (int r0 = 0; r0 < 16; ++r0)
      if ((r0 & tb) == 0) {
        const int r1 = r0 | tb;
        Cx a0 = s[r0], a1 = s[r1];
        s[r0].re = c ? a1.re : a0.re; s[r0].im = c ? a1.im : a0.im;
        s[r1].re = c ? a0.re : a1.re; s[r1].im = c ? a0.im : a1.im;
      }
  } else {
    constexpr int cm = 1 << (4 - CW), lm = 1 << (4 - TW);
    const bool c = (lane & cm) != 0;
#pragma unroll
    for (int r = 0; r < 16; ++r) {
      const float pre = sx(s[r].re, lm), pim = sx(s[r].im, lm);
      s[r].re = c ? pre : s[r].re;
      s[r].im = c ? pim : s[r].im;
    }
  }
}

template <int W>
static __device__ __forceinline__ void applyRY(Cx* s, float t, int lane) {
  const float c = __cosf(0.5f * t), sn = __sinf(0.5f * t);
  if constexpr (W >= 5) {
    constexpr int rb = 1 << (8 - W);
#pragma unroll
    for (int r0 = 0; r0 < 16; ++r0)
      if ((r0 & rb) == 0) {
        const int r1 = r0 | rb;
        Cx a0 = s[r0], a1 = s[r1];
        s[r0] = { fmaf(c, a0.re, -sn * a1.re), fmaf(c, a0.im, -sn * a1.im) };
        s[r1] = { fmaf(sn, a0.re, c * a1.re), fmaf(sn, a0.im, c * a1.im) };
      }
  } else {
    constexpr int lm = 1 << (4 - W);
    const bool hi = (lane & lm) != 0;
    const float A = hi ? sn : c;
    const float B = hi ? c : -sn;
#pragma unroll
    for (int r = 0; r < 16; ++r) {
      const float pre = sx(s[r].re, lm), pim = sx(s[r].im, lm);
      const float lre = hi ? pre : s[r].re, lim = hi ? pim : s[r].im;
      const float hre = hi ? s[r].re : pre, him = hi ? s[r].im : pim;
      s[r] = { fmaf(A, lre, B * hre), fmaf(A, lim, B * him) };
    }
  }
}

template <int W>
static __device__ __forceinline__ void applyRZ(Cx* s, float t, int lane) {
  const float c = __cosf(0.5f * t), sn = __sinf(0.5f * t);
  const Cx e0 = { c, -sn }, e1 = { c, sn };
  if constexpr (W >= 5) {
    constexpr int rb = 1 << (8 - W);
#pragma unroll
    for (int r = 0; r < 16; ++r) s[r] = cmul(s[r], (r & rb) ? e1 : e0);
  } else {
    constexpr int lm = 1 << (4 - W);
    const Cx ph = (lane & lm) ? e1 : e0;
#pragma unroll
    for (int r = 0; r < 16; ++r) s[r] = cmul(s[r], ph);
  }
}

// Rot(phi,theta,omega) = RZ(omega) RY(theta) RZ(phi) -- angles bounded, native trig.
static __device__ __forceinline__ U2 rotU(float phi, float th, float om) {
  const float c = __cosf(0.5f * th), sn = __sinf(0.5f * th);
  const float ap = 0.5f * (phi + om), am = 0.5f * (phi - om);
  const float ca = __cosf(ap), sa = __sinf(ap);
  const float cm = __cosf(am), sm = __sinf(am);
  U2 U;
  U.u00 = { c * ca, -c * sa };
  U.u01 = { -sn * cm, -sn * sm };
  U.u10 = { sn * cm, -sn * sm };
  U.u11 = { c * ca, c * sa };
  return U;
}

template <int W0, int W1, int W2>
static __device__ __forceinline__ void sel3(Cx* s, const U2& g0, const U2& g1, const U2& g2,
                                            int lane) {
  apply1<W0>(s, g0, lane);
  apply1<W1>(s, g1, lane);
  apply1<W2>(s, g2, lane);
  cnotg<W0, W1>(s, lane);
  cnotg<W1, W2>(s, lane);
  cnotg<W2, W0>(s, lane);
}

// One entangle iteration; all gate matrices precomputed (weight-derived,
// identical across the three iterations -- only wires differ).
template <int E, int NB>
static __device__ __forceinline__ void entangleStep(Cx* s, const U2* cg, const U2* s0g,
                                                    const U2* s1g, int lane) {
  applyC<NB, 7>(s, cg[0], lane);  // ctrl RX (nb -> a1)
  applyC<E, 7>(s, cg[1], lane);   // ctrl RY (e  -> a1)
  applyC<NB, 8>(s, cg[2], lane);  // ctrl RZ (nb -> a2)
  applyC<E, 8>(s, cg[3], lane);   // ctrl RY (e  -> a2)
  sel3<E, NB, 7>(s, s0g[0], s0g[1], s0g[2], lane);  // strong[0]
  sel3<7, NB, 8>(s, s1g[0], s1g[1], s1g[2], lane);  // strong[1]
}

__global__ void __launch_bounds__(256)
pqc_kernel(const float* __restrict__ x, const float* __restrict__ ebuf,
           float* __restrict__ x_acc,
           const int* __restrict__ subgraphs, const int* __restrict__ edge_ids,
           const float* __restrict__ strong, const float* __restrict__ inits,
           const float* __restrict__ update,
           const float* __restrict__ Wu1, const float* __restrict__ bu1,
           const float* __restrict__ Wu2, const float* __restrict__ bu2) {
  const int lane = threadIdx.x & 31;
  const int wave = threadIdx.x >> 5;
  const int node = blockIdx.x * (blockDim.x >> 5) + wave;

  float ang[7][2];
#pragma unroll
  for (int i = 0; i < 3; ++i) {
    const int ei = edge_ids[node * 3 + i];
    ang[i][0] = ebuf[ei * 2];
    ang[i][1] = ebuf[ei * 2 + 1];
  }
#pragma unroll
  for (int i = 0; i < 4; ++i) {
    const int si = subgraphs[node * 4 + i];
    ang[3 + i][0] = x[si * 2];
    ang[3 + i][1] = x[si * 2 + 1];
  }

  // ---- precompute all weight-derived gate matrices (uniform across wave) ----
  U2 cg[4];
  {
    float c, sn;
    c = __cosf(0.5f * inits[0]); sn = __sinf(0.5f * inits[0]);  // RX
    cg[0] = { { c, 0 }, { 0, -sn }, { 0, -sn }, { c, 0 } };
    c = __cosf(0.5f * inits[1]); sn = __sinf(0.5f * inits[1]);  // RY
    cg[1] = { { c, 0 }, { -sn, 0 }, { sn, 0 }, { c, 0 } };
    c = __cosf(0.5f * inits[2]); sn = __sinf(0.5f * inits[2]);  // RZ
    cg[2] = { { c, -sn }, { 0, 0 }, { 0, 0 }, { c, sn } };
    c = __cosf(0.5f * inits[3]); sn = __sinf(0.5f * inits[3]);  // RY
    cg[3] = { { c, 0 }, { -sn, 0 }, { sn, 0 }, { c, 0 } };
  }
  U2 s0g[3], s1g[3], upg[3];
#pragma unroll
  for (int j = 0; j < 3; ++j) {
    s0g[j] = rotU(strong[j * 3 + 0], strong[j * 3 + 1], strong[j * 3 + 2]);
    s1g[j] = rotU(strong[9 + j * 3 + 0], strong[9 + j * 3 + 1], strong[9 + j * 3 + 2]);
    upg[j] = rotU(update[j * 3 + 0], update[j * 3 + 1], update[j * 3 + 2]);
  }

  Cx s[16];
#pragma unroll
  for (int r = 0; r < 16; ++r) s[r] = { 0.f, 0.f };
  s[0].re = (lane == 0) ? 1.f : 0.f;  // |0...0>

  // encoding (RY, RZ on wires 0..6) with per-node angles
  applyRY<0>(s, ang[0][0], lane); applyRZ<0>(s, ang[0][1], lane);
  applyRY<1>(s, ang[1][0], lane); applyRZ<1>(s, ang[1][1], lane);
  applyRY<2>(s, ang[2][0], lane); applyRZ<2>(s, ang[2][1], lane);
  applyRY<3>(s, ang[3][0], lane); applyRZ<3>(s, ang[3][1], lane);
  applyRY<4>(s, ang[4][0], lane); applyRZ<4>(s, ang[4][1], lane);
  applyRY<5>(s, ang[5][0], lane); applyRZ<5>(s, ang[5][1], lane);
  applyRY<6>(s, ang[6][0], lane); applyRZ<6>(s, ang[6][1], lane);

  entangleStep<0, 4>(s, cg, s0g, s1g, lane);
  entangleStep<1, 5>(s, cg, s0g, s1g, lane);
  entangleStep<2, 6>(s, cg, s0g, s1g, lane);
  sel3<3, 7, 8>(s, upg[0], upg[1], upg[2], lane);

  // <X> on wire 3 (lane bit 1): summing over all lanes counts each pair twice.
  float acc = 0.f;
#pragma unroll
  for (int r = 0; r < 16; ++r) {
    const float pre = sx(s[r].re, 2), pim = sx(s[r].im, 2);
    acc = fmaf(s[r].re, pre, acc);
    acc = fmaf(s[r].im, pim, acc);
  }
#pragma unroll
  for (int m = 1; m < 32; m <<= 1) acc += sx(acc, m);

  // update MLP: [x[center], aggr] (3) -> 128 -> 2 ; 4 hidden units per lane
  const int center = subgraphs[node * 4 + 0];
  const float in0 = x[center * 2], in1 = x[center * 2 + 1], in2 = acc;
  float a0 = 0.f, a1 = 0.f;
#pragma unroll
  for (int k = 0; k < 4; ++k) {
    const int j = lane * 4 + k;
    float h = fmaf(in0, Wu1[j], fmaf(in1, Wu1[128 + j], fmaf(in2, Wu1[256 + j], bu1[j])));
    h = LRELU(h);
    a0 = fmaf(h, Wu2[j * 2 + 0], a0);
    a1 = fmaf(h, Wu2[j * 2 + 1], a1);
  }
#pragma unroll
  for (int m = 1; m < 32; m <<= 1) { a0 += sx(a0, m); a1 += sx(a1, m); }
  if (lane == 0) {
    atomicAdd(&x_acc[center * 2 + 0], a0 + bu2[0]);
    atomicAdd(&x_acc[center * 2 + 1], a1 + bu2[1]);
  }
}

// ======================= pool + head =======================================
__global__ void pool_head_kernel(const float* __restrict__ xf, const int* __restrict__ batch,
                                 const float* __restrict__ Wh1, const float* __restrict__ bh1,
                                 const float* __restrict__ Wh2, const float* __restrict__ bh2,
                                 float* __restrict__ out, int nnodes, int ngraphs) {
  __shared__ float gsum[32];
  __shared__ float gcnt[16];
  const int t = threadIdx.x;
  if (t < 2 * ngraphs) gsum[t] = 0.f;
  if (t < ngraphs) gcnt[t] = 0.f;
  __syncthreads();
  for (int n = t; n < nnodes; n += blockDim.x) {
    const int g = batch[n];
    atomicAdd(&gsum[2 * g + 0], xf[2 * n + 0]);
    atomicAdd(&gsum[2 * g + 1], xf[2 * n + 1]);
    atomicAdd(&gcnt[g], 1.f);
  }
  __syncthreads();
  if (t < ngraphs) {
    const float inv = 1.f / gcnt[t];
    const float g0 = gsum[2 * t] * inv, g1 = gsum[2 * t + 1] * inv;
    const float h0 = LRELU(fmaf(g0, Wh1[0], fmaf(g1, Wh1[2], bh1[0])));
    const float h1 = LRELU(fmaf(g0, Wh1[1], fmaf(g1, Wh1[3], bh1[1])));
    out[2 * t + 0] = fmaf(h0, Wh2[0], fmaf(h1, Wh2[2], bh2[0]));
    out[2 * t + 1] = fmaf(h0, Wh2[1], fmaf(h1, Wh2[3], bh2[1]));
  }
}

// ======================= launcher ==========================================
extern "C" void kernel_launch(void* const* d_in, const int* in_sizes, int n_in,
                              void* d_out, int out_size, void* d_ws, size_t ws_size,
                              hipStream_t stream) {
  const float* node_feat = (const float*)d_in[0];
  const float* edge_attr = (const float*)d_in[1];
  const float* Wn1 = (const float*)d_in[2];  const float* bn1 = (const float*)d_in[3];
  const float* Wn2 = (const float*)d_in[4];  const float* bn2 = (const float*)d_in[5];
  const float* We1 = (const float*)d_in[6];  const float* be1 = (const float*)d_in[7];
  const float* We2 = (const float*)d_in[8];  const float* be2 = (const float*)d_in[9];
  const float* strong = (const float*)d_in[10];
  const float* inits  = (const float*)d_in[11];
  const float* update = (const float*)d_in[12];
  const float* Wu1 = (const float*)d_in[13]; const float* bu1 = (const float*)d_in[14];
  const float* Wu2 = (const float*)d_in[15]; const float* bu2 = (const float*)d_in[16];
  const float* Wh1 = (const float*)d_in[17]; const float* bh1 = (const float*)d_in[18];
  const float* Wh2 = (const float*)d_in[19]; const float* bh2 = (const float*)d_in[20];
  const int* subgraphs = (const int*)d_in[21];
  const int* edge_ids  = (const int*)d_in[22];
  const int* batch     = (const int*)d_in[23];

  float* ws    = (float*)d_ws;
  float* x     = ws;              // 4096*2
  float* x_acc = ws + 8192;       // 4096*2
  float* ebuf  = ws + 16384;      // 12288*2

  feat_mlp_kernel<8><<<64, 128, 0, stream>>>(node_feat, Wn1, bn1, Wn2, bn2, x, x_acc, 4096);
  feat_mlp_kernel<4><<<192, 128, 0, stream>>>(edge_attr, We1, be1, We2, be2, ebuf, nullptr, 12288);
  pqc_kernel<<<512, 256, 0, stream>>>(x, ebuf, x_acc, subgraphs, edge_ids,
                                      strong, inits, update, Wu1, bu1, Wu2, bu2);
  pool_head_kernel<<<1, 256, 0, stream>>>(x_acc, batch, Wh1, bh1, Wh2, bh2,
                                          (float*)d_out, 4096, 16);
}